// AnswerPredictor_49572512530438
// MI455X (gfx1250) — compile-verified
//
#include <hip/hip_runtime.h>

typedef __attribute__((ext_vector_type(2))) float v2f;
typedef __attribute__((ext_vector_type(8))) float v8f;

#define DIM   384
#define BATCH 1024
#define DD    (DIM * DIM)
#define NTILE (DIM / 16)          // 24 column tiles
#define MTILE (BATCH / 16)        // 64 row tiles

// ---------------------------------------------------------------------------
// Prep 1: Pf[i*DIM+j] = rw[i] * (1-W0[i,j])*(1-W1[i,j])*(1-W2[i,j])
// ---------------------------------------------------------------------------
__global__ __launch_bounds__(256) void prep_P(const float* __restrict__ W,
                                              const float* __restrict__ rw,
                                              float* __restrict__ Pf) {
    int idx = blockIdx.x * 256 + threadIdx.x;   // 0 .. DD-1 exactly
    int i   = idx / DIM;
    float w0 = W[idx], w1 = W[DD + idx], w2 = W[2 * DD + idx];
    Pf[idx] = rw[i] * ((1.f - w0) * (1.f - w1) * (1.f - w2));
}

// ---------------------------------------------------------------------------
// Prep 2: c0[j] = sum_i rw[i] * ((b0*(1-w1)+b1)*(1-w2)+b2)[i,j]
// Fixed-order tree reduction -> deterministic.
// ---------------------------------------------------------------------------
__global__ __launch_bounds__(128) void reduce_c0(const float* __restrict__ W,
                                                 const float* __restrict__ bb,
                                                 const float* __restrict__ rw,
                                                 float* __restrict__ c0) {
    __shared__ float s[128];
    int j = blockIdx.x;
    float acc = 0.f;
    for (int i = threadIdx.x; i < DIM; i += 128) {
        int idx = i * DIM + j;
        float w1 = W[DD + idx], w2 = W[2 * DD + idx];
        float b0 = bb[idx], b1 = bb[DD + idx], b2 = bb[2 * DD + idx];
        acc += rw[i] * ((b0 * (1.f - w1) + b1) * (1.f - w2) + b2);
    }
    s[threadIdx.x] = acc;
    __syncthreads();
    for (int off = 64; off > 0; off >>= 1) {
        if (threadIdx.x < off) s[threadIdx.x] += s[threadIdx.x + off];
        __syncthreads();
    }
    if (threadIdx.x == 0) c0[j] = s[0];
}

// ---------------------------------------------------------------------------
// GEMM1: pooled[b,j] = v2[b,j] * (v1 @ Pf)[b,j] - c0[j]
// A = v1 (1024x384 row-major), B[k][n] = Pf[k*DIM+n] (row-major, no transpose)
// One wave per 16x16 tile, V_WMMA_F32_16X16X4_F32, K = 384 in steps of 4.
// ---------------------------------------------------------------------------
__global__ __launch_bounds__(256) void gemm1_pool(const float* __restrict__ A,
                                                  const float* __restrict__ Bm,
                                                  const float* __restrict__ V2,
                                                  const float* __restrict__ c0,
                                                  float* __restrict__ pooled) {
    const int wave = threadIdx.x >> 5;
    const int lane = threadIdx.x & 31;
    const int lm   = lane & 15;       // row (A) / col (B,C) within tile
    const int hi   = lane >> 4;       // K-half selector
    const int tile = blockIdx.x * 8 + wave;          // 0..1535, exact
    const int m0   = (tile / NTILE) * 16;
    const int n0   = (tile % NTILE) * 16;

    const float* arow = A + (m0 + lm) * DIM;         // this lane's A row
    const float* bcol = Bm + n0 + lm;                // this lane's B column

    v8f acc = {};
    #pragma unroll 8
    for (int k = 0; k < DIM; k += 4) {
        // A 16x4: VGPR0 = K 2*hi, VGPR1 = K 2*hi+1 (contiguous pair)
        v2f a = *(const v2f*)(arow + k + 2 * hi);
        // B 4x16: same K striping, column-strided in memory
        v2f b;
        b.x = bcol[(k + 2 * hi) * DIM];
        b.y = bcol[(k + 2 * hi + 1) * DIM];
        acc = __builtin_amdgcn_wmma_f32_16x16x4_f32(
            /*neg_a=*/false, a, /*neg_b=*/false, b,
            /*c_mod=*/(short)0, acc, /*reuse_a=*/false, /*reuse_b=*/false);
    }

    // Fused epilogue: pooled = v2 * t - c0
    const int col = n0 + lm;
    const float c0v = c0[col];
    #pragma unroll
    for (int r = 0; r < 8; ++r) {
        const int row = m0 + r + hi * 8;
        pooled[row * DIM + col] = V2[row * DIM + col] * acc[r] - c0v;
    }
}

// ---------------------------------------------------------------------------
// GEMM2: out = pooled @ lin_W^T + lin_b
// B[k][n] = lin_W[n*DIM + k]  -> per-lane contiguous float2 along k
// ---------------------------------------------------------------------------
__global__ __launch_bounds__(256) void gemm2_out(const float* __restrict__ A,
                                                 const float* __restrict__ LW,
                                                 const float* __restrict__ lb,
                                                 float* __restrict__ out) {
    const int wave = threadIdx.x >> 5;
    const int lane = threadIdx.x & 31;
    const int lm   = lane & 15;
    const int hi   = lane >> 4;
    const int tile = blockIdx.x * 8 + wave;
    const int m0   = (tile / NTILE) * 16;
    const int n0   = (tile % NTILE) * 16;

    const float* arow = A + (m0 + lm) * DIM;
    const float* brow = LW + (n0 + lm) * DIM;        // lin_W row n -> B column n

    v8f acc = {};
    #pragma unroll 8
    for (int k = 0; k < DIM; k += 4) {
        v2f a = *(const v2f*)(arow + k + 2 * hi);
        v2f b = *(const v2f*)(brow + k + 2 * hi);
        acc = __builtin_amdgcn_wmma_f32_16x16x4_f32(
            false, a, false, b, (short)0, acc, false, false);
    }

    const int col = n0 + lm;
    const float bias = lb[col];
    #pragma unroll
    for (int r = 0; r < 8; ++r) {
        const int row = m0 + r + hi * 8;
        out[row * DIM + col] = acc[r] + bias;
    }
}

// ---------------------------------------------------------------------------
// Launch
// ---------------------------------------------------------------------------
extern "C" void kernel_launch(void* const* d_in, const int* in_sizes, int n_in,
                              void* d_out, int out_size, void* d_ws, size_t ws_size,
                              hipStream_t stream) {
    const float* v1  = (const float*)d_in[0];   // (1024, 384)
    const float* v2  = (const float*)d_in[1];   // (1024, 384)
    const float* bW  = (const float*)d_in[2];   // (3, 384, 384)
    const float* bb  = (const float*)d_in[3];   // (3, 384, 384)
    const float* rw  = (const float*)d_in[4];   // (384,)
    const float* lW  = (const float*)d_in[5];   // (384, 384)
    const float* lb  = (const float*)d_in[6];   // (384,)
    float* out = (float*)d_out;                 // (1024, 384)

    float* Pf     = (float*)d_ws;               // DD floats
    float* c0     = Pf + DD;                    // DIM floats
    float* pooled = c0 + DIM;                   // BATCH*DIM floats

    prep_P<<<DD / 256, 256, 0, stream>>>(bW, rw, Pf);
    reduce_c0<<<DIM, 128, 0, stream>>>(bW, bb, rw, c0);

    const int gemm_blocks = (MTILE * NTILE) / 8;   // 1536 tiles / 8 waves = 192
    gemm1_pool<<<gemm_blocks, 256, 0, stream>>>(v1, Pf, v2, c0, pooled);
    gemm2_out<<<gemm_blocks, 256, 0, stream>>>(pooled, lW, lb, out);
}